// HDN_conv_block_14061722927150
// MI455X (gfx1250) — compile-verified
//
#include <hip/hip_runtime.h>
#include <math.h>

// Problem constants (match reference)
#define CCH  200     // channels = H*D
#define HNUM 4       // heads
#define DD   50      // per-head dim
#define GNUM 512     // graphs
#define NT   13      // col tiles of 16 covering 200 (last partial)

typedef __attribute__((ext_vector_type(2))) float v2f;
typedef __attribute__((ext_vector_type(8))) float v8f;
typedef __attribute__((ext_vector_type(4))) unsigned int v4u;
typedef __attribute__((ext_vector_type(4))) int v4i;
typedef __attribute__((ext_vector_type(8))) int v8i;

// ---- ordered-int encoding for float atomicMax (monotone map) ----
__device__ __forceinline__ int f2ord(float f) {
  int b = __float_as_int(f);
  return b >= 0 ? b : (b ^ 0x7fffffff);
}
__device__ __forceinline__ float ord2f(int k) {
  return __int_as_float(k >= 0 ? k : (k ^ 0x7fffffff));
}

// wave32 butterfly sum
__device__ __forceinline__ float wsum(float v) {
#pragma unroll
  for (int o = 16; o > 0; o >>= 1) v += __shfl_xor(v, o, 32);
  return v;
}

// ---------------------------------------------------------------------------
// init: per-graph stats, per-(node,head) softmax state
// ---------------------------------------------------------------------------
__global__ __launch_bounds__(256)
void k_init(float* s_sum, float* s_sq, float* s_cnt, float* gsum, int* gmax_enc,
            int* m_enc, float* z, int nh) {
  int i = blockIdx.x * blockDim.x + threadIdx.x;
  if (i < nh) { m_enc[i] = 0x80000000; z[i] = 0.0f; }
  if (i < GNUM) {
    s_sum[i] = 0.0f; s_sq[i] = 0.0f; s_cnt[i] = 0.0f;
    gsum[i] = 0.0f; gmax_enc[i] = 0x80000000;
  }
}

// ---------------------------------------------------------------------------
// LayerNorm(graph mode) stats: one wave per node
// ---------------------------------------------------------------------------
__global__ __launch_bounds__(256)
void k_ln_stats(const float* __restrict__ x, const int* __restrict__ batch,
                float* s_sum, float* s_sq, float* s_cnt, int n) {
  int w = (blockIdx.x * blockDim.x + threadIdx.x) >> 5;
  int lane = threadIdx.x & 31;
  if (w >= n) return;
  const float* xr = x + (size_t)w * CCH;
  float s = 0.0f, q = 0.0f;
  for (int c = lane; c < CCH; c += 32) { float v = xr[c]; s += v; q += v * v; }
  s = wsum(s); q = wsum(q);
  if (lane == 0) {
    int g = batch[w];
    atomicAdd(&s_sum[g], s);
    atomicAdd(&s_sq[g], q);
    atomicAdd(&s_cnt[g], 1.0f);
  }
}

__global__ __launch_bounds__(256)
void k_ln_final(const float* s_sum, const float* s_sq, const float* s_cnt,
                float* mean, float* rstd) {
  int g = blockIdx.x * blockDim.x + threadIdx.x;
  if (g >= GNUM) return;
  float den = fmaxf(s_cnt[g] * (float)CCH, 1.0f);
  float m = s_sum[g] / den;
  float v = s_sq[g] / den - m * m;
  mean[g] = m;
  rstd[g] = rsqrtf(v + 1e-5f);
}

__global__ __launch_bounds__(256)
void k_ln_apply(const float* __restrict__ x, const int* __restrict__ batch,
                const float* mean, const float* rstd,
                const float* lw, const float* lb, float* __restrict__ h, int n) {
  size_t i = (size_t)blockIdx.x * blockDim.x + threadIdx.x;
  if (i >= (size_t)n * CCH) return;
  int node = (int)(i / CCH), c = (int)(i % CCH);
  int g = batch[node];
  float v = (x[i] - mean[g]) * rstd[g] * lw[c] + lb[c];
  h[i] = v > 0.0f ? v : expm1f(v);               // ELU
}

// ---------------------------------------------------------------------------
// GEMM: xh = h @ W   via V_WMMA_F32_16X16X4_F32.
// - Whole W (200x200 f32 = 160KB) staged into LDS once per block by the
//   Tensor Data Mover (tensor_load_to_lds + s_wait_tensorcnt).
// - Each wave owns a 16-row strip and ALL 13 column tiles, so A is streamed
//   from HBM exactly once (roofline-minimal traffic).
// - Compiler barriers force the LDS reads to be materialized (the TDM write
//   to LDS is invisible to LLVM's alias analysis).
// ---------------------------------------------------------------------------
__global__ __launch_bounds__(256)
void k_gemm_wmma(const float* __restrict__ A, const float* __restrict__ B,
                 float* __restrict__ Dst, int nrows) {
  __shared__ float ldsW[CCH * CCH];     // 160,000 B (<= 320KB WGP LDS)

  // Never taken at runtime (nrows > 0), but guarantees a visible store to
  // ldsW so the compiler cannot undef-fold its loads.
  if (nrows < 0) ldsW[threadIdx.x] = B[threadIdx.x];

  // ---- TDM: DMA the full weight matrix into LDS (wave 0 issues) ----
  if (threadIdx.x < 32) {
    unsigned long long wa = (unsigned long long)(const void*)B;
    v4u g0;
    g0.x = 1u;                                   // count = 1 valid descriptor
    g0.y = 0u;                                   // lds_addr = 0 (ldsW base)
    g0.z = (unsigned int)(wa & 0xffffffffu);     // global_addr[31:0]
    g0.w = (unsigned int)((wa >> 32) & 0x1ffffffu) | (2u << 30); // addr[56:32] | type=2
    v8i g1;
    g1[0] = (int)(2u << 16);                     // data_size = 2 (4 bytes)
    g1[1] = (int)((unsigned)CCH << 16);          // tensor_dim0[15:0] @ bits 63:48
    g1[2] = (int)((unsigned)CCH << 16);          // tensor_dim1[15:0] @ bits 95:80
    g1[3] = (int)((unsigned)CCH << 16);          // tile_dim0 @ bits 127:112
    g1[4] = (int)CCH;                            // tile_dim1 @ bits 143:128
    g1[5] = (int)CCH;                            // tensor_dim0_stride[31:0]
    g1[6] = 0;
    g1[7] = 0;
    v4i z4 = {};
#if __clang_major__ >= 23
    v8i z8 = {};
    __builtin_amdgcn_tensor_load_to_lds(g0, g1, z4, z4, z8, 0);
#else
    __builtin_amdgcn_tensor_load_to_lds(g0, g1, z4, z4, 0);
#endif
    __builtin_amdgcn_s_wait_tensorcnt(0);
    asm volatile("" ::: "memory");               // TDM wrote LDS: tell LLVM
  }
  __syncthreads();
  asm volatile("" ::: "memory");                 // keep ldsW reads below here

  const int wave = threadIdx.x >> 5;
  const int m0 = (blockIdx.x * 8 + wave) * 16;   // 16-row strip per wave
  if (m0 >= nrows) return;                        // wave-uniform, after barrier

  const int lane = threadIdx.x & 31;
  const int half = lane >> 4;
  const int l    = lane & 15;
  const int row  = m0 + l;
  const float* arow = A + (size_t)row * CCH + 2 * half;
  __builtin_prefetch(arow, 0, 0);
  __builtin_prefetch(arow + 128, 0, 0);

  v8f zero = {};
  v8f acc[NT];
#pragma unroll
  for (int t = 0; t < NT; ++t) acc[t] = zero;

  for (int k = 0; k < CCH; k += 4) {
    v2f a = *(const v2f*)(arow + k);             // h[row][k+2h], h[row][k+2h+1]
    const float* wr0 = &ldsW[(k + 2 * half) * CCH];
    const float* wr1 = wr0 + CCH;
#pragma unroll
    for (int t = 0; t < NT; ++t) {
      const int col = t * 16 + l;
      v2f b;
      if (t < NT - 1) {
        b.x = wr0[col];                          // W[k+2h  ][col]
        b.y = wr1[col];                          // W[k+2h+1][col]
      } else {                                   // partial tile: clamp + mask
        const float mask = (col < CCH) ? 1.0f : 0.0f;
        const int cc = (col < CCH) ? col : 0;
        b.x = wr0[cc] * mask;
        b.y = wr1[cc] * mask;
      }
      acc[t] = __builtin_amdgcn_wmma_f32_16x16x4_f32(false, a, false, b,
                                                     (short)0, acc[t], false, false);
    }
  }

  const int dr = m0 + half * 8;                  // D layout: M = v + 8*half
#pragma unroll
  for (int t = 0; t < NT; ++t) {
    const int col = t * 16 + l;
    if (col < CCH) {
#pragma unroll
      for (int v = 0; v < 8; ++v)
        Dst[(size_t)(dr + v) * CCH + col] = acc[t][v];
    }
  }
}

// ---------------------------------------------------------------------------
// a_src/a_dst: per-node, per-head dot with attention vectors (wave per node)
// note att_src flat index == channel index (c = head*50 + d)
// ---------------------------------------------------------------------------
__global__ __launch_bounds__(256)
void k_att(const float* __restrict__ xh, const float* __restrict__ as,
           const float* __restrict__ ad, float* a_src, float* a_dst, int n) {
  int w = (blockIdx.x * blockDim.x + threadIdx.x) >> 5;
  int lane = threadIdx.x & 31;
  if (w >= n) return;
  const float* xr = xh + (size_t)w * CCH;
  float ps[HNUM] = {0, 0, 0, 0}, pd[HNUM] = {0, 0, 0, 0};
  for (int c = lane; c < CCH; c += 32) {
    float v = xr[c];
    int hh = c / DD;
    ps[hh] += v * as[c];
    pd[hh] += v * ad[c];
  }
#pragma unroll
  for (int hh = 0; hh < HNUM; ++hh) {
    float s = wsum(ps[hh]), d = wsum(pd[hh]);
    if (lane == 0) { a_src[w * HNUM + hh] = s; a_dst[w * HNUM + hh] = d; }
  }
}

__device__ __forceinline__ void edge_sd(const int* ei, int E, int idx, int& s, int& d) {
  if (idx < E) { s = ei[idx]; d = ei[E + idx]; } else { s = d = idx - E; }
}

// ---------------------------------------------------------------------------
// segment softmax over incoming edges: max pass, sum pass, scatter pass
// ---------------------------------------------------------------------------
__global__ __launch_bounds__(256)
void k_edge_max(const int* __restrict__ ei, const float* __restrict__ a_src,
                const float* __restrict__ a_dst, int* m_enc, int E, int ET) {
  int idx = blockIdx.x * blockDim.x + threadIdx.x;
  if (idx >= ET) return;
  int s, d; edge_sd(ei, E, idx, s, d);
#pragma unroll
  for (int hh = 0; hh < HNUM; ++hh) {
    float al = a_src[s * HNUM + hh] + a_dst[d * HNUM + hh];
    al = al > 0.0f ? al : 0.2f * al;              // leaky_relu
    atomicMax(&m_enc[d * HNUM + hh], f2ord(al));
  }
}

__global__ __launch_bounds__(256)
void k_edge_sum(const int* __restrict__ ei, const float* __restrict__ a_src,
                const float* __restrict__ a_dst, const int* __restrict__ m_enc,
                float* z, int E, int ET) {
  int idx = blockIdx.x * blockDim.x + threadIdx.x;
  if (idx >= ET) return;
  int s, d; edge_sd(ei, E, idx, s, d);
#pragma unroll
  for (int hh = 0; hh < HNUM; ++hh) {
    float al = a_src[s * HNUM + hh] + a_dst[d * HNUM + hh];
    al = al > 0.0f ? al : 0.2f * al;
    atomicAdd(&z[d * HNUM + hh], expf(al - ord2f(m_enc[d * HNUM + hh])));
  }
}

// out[dst] += xh[src] * att   (wave per edge; 200 coalesced atomic f32 adds)
__global__ __launch_bounds__(256)
void k_edge_scatter(const int* __restrict__ ei, const float* __restrict__ xh,
                    const float* __restrict__ a_src, const float* __restrict__ a_dst,
                    const int* __restrict__ m_enc, const float* __restrict__ z,
                    float* out, int E, int ET) {
  int w = (blockIdx.x * blockDim.x + threadIdx.x) >> 5;
  int lane = threadIdx.x & 31;
  if (w >= ET) return;
  int s, d; edge_sd(ei, E, w, s, d);
  float att[HNUM];
#pragma unroll
  for (int hh = 0; hh < HNUM; ++hh) {
    float al = a_src[s * HNUM + hh] + a_dst[d * HNUM + hh];
    al = al > 0.0f ? al : 0.2f * al;
    att[hh] = expf(al - ord2f(m_enc[d * HNUM + hh])) / (z[d * HNUM + hh] + 1e-16f);
  }
  const float* xr = xh + (size_t)s * CCH;
  float* orow = out + (size_t)d * CCH;
  for (int c = lane; c < CCH; c += 32)
    atomicAdd(&orow[c], xr[c] * att[c / DD]);
}

__global__ __launch_bounds__(256)
void k_bias(float* out, const float* __restrict__ bias, int n) {
  size_t i = (size_t)blockIdx.x * blockDim.x + threadIdx.x;
  if (i >= (size_t)n * CCH) return;
  out[i] += bias[i % CCH];
}

// t[i] = out[i].w_rel ; sc[i] = out[i].w_root + b_rel   (wave per node)
__global__ __launch_bounds__(256)
void k_dots(const float* __restrict__ out, const float* __restrict__ w_rel,
            const float* __restrict__ w_root, const float* __restrict__ b_rel,
            float* t, float* sc, int n) {
  int w = (blockIdx.x * blockDim.x + threadIdx.x) >> 5;
  int lane = threadIdx.x & 31;
  if (w >= n) return;
  const float* orow = out + (size_t)w * CCH;
  float tr = 0.0f, ro = 0.0f;
  for (int c = lane; c < CCH; c += 32) {
    float v = orow[c];
    tr += v * w_rel[c];
    ro += v * w_root[c];
  }
  tr = wsum(tr); ro = wsum(ro);
  if (lane == 0) { t[w] = tr; sc[w] = ro + b_rel[0]; }
}

// sc[dst] += t[src] over the E real edges (no self loops)
__global__ __launch_bounds__(256)
void k_score_scatter(const int* __restrict__ ei, const float* __restrict__ t,
                     float* sc, int E) {
  int idx = blockIdx.x * blockDim.x + threadIdx.x;
  if (idx >= E) return;
  atomicAdd(&sc[ei[E + idx]], t[ei[idx]]);
}

__global__ __launch_bounds__(256)
void k_gmax(const float* __restrict__ sc, const int* __restrict__ batch,
            int* gmax_enc, int n) {
  int i = blockIdx.x * blockDim.x + threadIdx.x;
  if (i >= n) return;
  atomicMax(&gmax_enc[batch[i]], f2ord(sc[i]));
}

__global__ __launch_bounds__(256)
void k_gexp(const float* __restrict__ sc, const int* __restrict__ batch,
            const int* __restrict__ gmax_enc, float* es, float* gsum, int n) {
  int i = blockIdx.x * blockDim.x + threadIdx.x;
  if (i >= n) return;
  float e = expf(sc[i] - ord2f(gmax_enc[batch[i]]));
  es[i] = e;
  atomicAdd(&gsum[batch[i]], e);
}

// xp = out * softmax_score ; g[batch] += xp  (wave per node)
__global__ __launch_bounds__(256)
void k_final(const float* __restrict__ out, const int* __restrict__ batch,
             const float* __restrict__ es, const float* __restrict__ gsum,
             float* __restrict__ xp, float* __restrict__ gout, int n) {
  int w = (blockIdx.x * blockDim.x + threadIdx.x) >> 5;
  int lane = threadIdx.x & 31;
  if (w >= n) return;
  int g = batch[w];
  float s = es[w] / gsum[g];
  const float* orow = out + (size_t)w * CCH;
  float* xr = xp + (size_t)w * CCH;
  float* gr = gout + (size_t)g * CCH;
  for (int c = lane; c < CCH; c += 32) {
    float v = orow[c] * s;
    xr[c] = v;
    atomicAdd(&gr[c], v);
  }
}

// ---------------------------------------------------------------------------
extern "C" void kernel_launch(void* const* d_in, const int* in_sizes, int n_in,
                              void* d_out, int out_size, void* d_ws, size_t ws_size,
                              hipStream_t stream) {
  const float* x      = (const float*)d_in[0];
  const int*   ei     = (const int*)d_in[1];   // [2*E] flat: row0=src, row1=dst
  const int*   batch  = (const int*)d_in[2];
  const float* ln_w   = (const float*)d_in[3];
  const float* ln_b   = (const float*)d_in[4];
  const float* W_gat  = (const float*)d_in[5]; // [C, H*D] row-major
  const float* att_s  = (const float*)d_in[6]; // [H*D] flat
  const float* att_d  = (const float*)d_in[7];
  const float* bias_g = (const float*)d_in[8];
  const float* w_rel  = (const float*)d_in[9];
  const float* b_rel  = (const float*)d_in[10];
  const float* w_root = (const float*)d_in[11];

  const int N  = in_sizes[0] / CCH;
  const int E  = in_sizes[1] / 2;
  const int ET = E + N;           // edges incl. self loops
  const size_t NC = (size_t)N * CCH;
  const int NH = N * HNUM;

  // ---- workspace layout (floats) ----
  float* ws    = (float*)d_ws;
  float* hbuf  = ws;              // [N*C]  h, later reused as `out`
  float* xh    = ws + NC;         // [N*C]
  float* sm    = ws + 2 * NC;     // small region
  float* a_src = sm;                       // [N*H]
  float* a_dst = a_src + NH;               // [N*H]
  int*   m_enc = (int*)(a_dst + NH);       // [N*H]
  float* z     = (float*)(m_enc + NH);     // [N*H]
  float* t     = z + NH;                   // [N]
  float* sc    = t + N;                    // [N]
  float* es    = sc + N;                   // [N]
  float* s_sum = es + N;                   // [G]
  float* s_sq  = s_sum + GNUM;             // [G]
  float* s_cnt = s_sq + GNUM;              // [G]
  float* mean  = s_cnt + GNUM;             // [G]
  float* rstd  = mean + GNUM;              // [G]
  int*   gmax  = (int*)(rstd + GNUM);      // [G]
  float* gsum  = (float*)(gmax + GNUM);    // [G]

  float* xp_out = (float*)d_out;           // [N*C]
  float* g_out  = xp_out + NC;             // [G*C]

  const int TB = 256;
  auto cdiv = [](long long a, long long b) { return (int)((a + b - 1) / b); };

  // 0. init small state
  k_init<<<cdiv(NH, TB), TB, 0, stream>>>(s_sum, s_sq, s_cnt, gsum, gmax, m_enc, z, NH);
  // 1. LayerNorm (graph mode) + ELU
  k_ln_stats<<<cdiv(N, TB / 32), TB, 0, stream>>>(x, batch, s_sum, s_sq, s_cnt, N);
  k_ln_final<<<cdiv(GNUM, TB), TB, 0, stream>>>(s_sum, s_sq, s_cnt, mean, rstd);
  k_ln_apply<<<cdiv((long long)NC, TB), TB, 0, stream>>>(x, batch, mean, rstd, ln_w, ln_b, hbuf, N);
  // 2. GEMM on the WMMA pipe: xh = h @ W_gat (TDM-staged W, 16x208 per wave)
  k_gemm_wmma<<<cdiv(N, 128), TB, 0, stream>>>(hbuf, W_gat, xh, N);
  // 3. attention logits per node/head
  k_att<<<cdiv(N, TB / 32), TB, 0, stream>>>(xh, att_s, att_d, a_src, a_dst, N);
  // 4. reuse hbuf as `out`, zero it (h is dead after the GEMM)
  hipMemsetAsync(hbuf, 0, NC * sizeof(float), stream);
  float* out = hbuf;
  // 5. segment softmax + message scatter
  k_edge_max<<<cdiv(ET, TB), TB, 0, stream>>>(ei, a_src, a_dst, m_enc, E, ET);
  k_edge_sum<<<cdiv(ET, TB), TB, 0, stream>>>(ei, a_src, a_dst, m_enc, z, E, ET);
  k_edge_scatter<<<cdiv(ET, TB / 32), TB, 0, stream>>>(ei, xh, a_src, a_dst, m_enc, z, out, E, ET);
  k_bias<<<cdiv((long long)NC, TB), TB, 0, stream>>>(out, bias_g, N);
  // 6. SAGPooling score: GraphConv(C->1) then per-graph softmax
  k_dots<<<cdiv(N, TB / 32), TB, 0, stream>>>(out, w_rel, w_root, b_rel, t, sc, N);
  k_score_scatter<<<cdiv(E, TB), TB, 0, stream>>>(ei, t, sc, E);
  k_gmax<<<cdiv(N, TB), TB, 0, stream>>>(sc, batch, gmax, N);
  k_gexp<<<cdiv(N, TB), TB, 0, stream>>>(sc, batch, gmax, es, gsum, N);
  // 7. outputs: xp and pooled g (zero g region of d_out first)
  hipMemsetAsync(g_out, 0, (size_t)GNUM * CCH * sizeof(float), stream);
  k_final<<<cdiv(N, TB / 32), TB, 0, stream>>>(out, batch, es, gsum, xp_out, g_out, N);
}